// NGCF_DISCO_77790447665436
// MI455X (gfx1250) — compile-verified
//
#include <hip/hip_runtime.h>
#include <math.h>

#define USERS   100000
#define ITEMS   50000
#define NTOT    150000
#define NPAD    150016      // padded to multiple of 64 rows (64*2344)
#define DD      64
#define NNZE    2400000
#define BB      4096
#define NLVL    3
#define LEAF    32
#define HIDN    256
#define EPSV    0.1f
#define TAUV    0.2f

typedef float v2f __attribute__((ext_vector_type(2)));
typedef float v8f __attribute__((ext_vector_type(8)));

// ---- f32 WMMA 16x16x4 wrapper (D = A(16x4) * B(4x16) + C) ----
static __device__ __forceinline__ v8f wmma4(v2f a, v2f b, v8f c) {
  return __builtin_amdgcn_wmma_f32_16x16x4_f32(false, a, false, b, (short)0, c, false, false);
}

static __device__ __forceinline__ v2f ld2(const float* p) {
  return *(const v2f*)p;   // 8-byte aligned by construction -> b64 load
}

static __device__ __forceinline__ float wave_sum32(float v) {
  v += __shfl_xor(v, 1);  v += __shfl_xor(v, 2);  v += __shfl_xor(v, 4);
  v += __shfl_xor(v, 8);  v += __shfl_xor(v, 16);
  return v;
}

static __device__ __forceinline__ float hash01(unsigned x) {
  x ^= x >> 16; x *= 0x7feb352dU; x ^= x >> 15; x *= 0x846ca68bU; x ^= x >> 16;
  return (float)(x >> 8) * (1.0f / 16777216.0f);
}

// ---------------- init / zero ----------------
__global__ void k_zero(float* p, long n) {
  long i = (long)blockIdx.x * blockDim.x + threadIdx.x;
  if (i < n) p[i] = 0.f;
}

__global__ void k_init_E(const float* __restrict__ ego, float* E, float* Esum) {
  long i = (long)blockIdx.x * 256 + threadIdx.x;   // grid sized to exactly NPAD*DD
  float v = (i < (long)NTOT * DD) ? ego[i] : 0.f;
  E[i] = v; Esum[i] = v;
}

// ---------------- SpMM: side[row] += val * E[col]  (one wave per edge) ----------------
__global__ void k_spmm(const float* __restrict__ vals, const int* __restrict__ rows,
                       const int* __restrict__ cols, const float* __restrict__ E,
                       float* side) {
  int wid = (int)blockIdx.x * (blockDim.x >> 5) + (threadIdx.x >> 5);
  if (wid >= NNZE) return;
  int lane = threadIdx.x & 31;
  if (lane == 0) {  // stream-ahead hint for the nnz arrays (speculative, OOB-safe)
    __builtin_prefetch(vals + wid + 4096, 0, 1);
    __builtin_prefetch(rows + wid + 4096, 0, 1);
    __builtin_prefetch(cols + wid + 4096, 0, 1);
  }
  int c = cols[wid], r = rows[wid];
  float v = vals[wid];
  v2f ev = ld2(E + (size_t)c * DD + lane * 2);
  float* sp = side + (size_t)r * DD + lane * 2;
  atomicAdd(sp,     v * ev.x);
  atomicAdd(sp + 1, v * ev.y);
}

// ------- fused NGCF dense layer: E = l2norm(lrelu(side@gc+gcb)+lrelu((E*side)@bi+bib)); Esum += E -------
// block = 128 (4 waves), wave owns 16 rows, block owns 64 rows.
// Weights staged transposed in LDS so each B fragment is one ds_load_b64 at an immediate offset.
__global__ void __launch_bounds__(128) k_ngcf_dense(
    const float* __restrict__ side, float* E, float* Esum,
    const float* __restrict__ gcw, const float* __restrict__ gcb,
    const float* __restrict__ biw, const float* __restrict__ bib) {
  __shared__ float gT[64][68];           // gT[n][k] = gcw[k*64+n]; stride 68 -> conflict-free frags
  __shared__ float bT[64][68];
  __shared__ float outs[4][16][68];
  int tid = threadIdx.x;
  for (int i = tid; i < 4096; i += 128) {
    int k = i >> 6, n = i & 63;
    gT[n][k] = gcw[i];
    bT[n][k] = biw[i];
  }
  __syncthreads();

  int lane = tid & 31, wave = tid >> 5;
  int row0 = ((int)blockIdx.x * 4 + wave) * 16;
  int r = lane & 15, h = lane >> 4, kk = h * 2;
  const float* srow = side + (size_t)(row0 + r) * 64 + kk;
  const float* erow = E    + (size_t)(row0 + r) * 64 + kk;

  for (int n0 = 0; n0 < 64; n0 += 16) {
    v8f c1 = {}; v8f c2 = {};
    const float* gr = &gT[n0 + r][kk];
    const float* br = &bT[n0 + r][kk];
#pragma unroll
    for (int k0 = 0; k0 < 64; k0 += 4) {
      v2f s = ld2(srow + k0);
      v2f e = ld2(erow + k0);
      v2f p; p.x = s.x * e.x; p.y = s.y * e.y;
      v2f b1 = ld2(gr + k0);
      v2f b2 = ld2(br + k0);
      c1 = wmma4(s, b1, c1);
      c2 = wmma4(p, b2, c2);
    }
    float g = gcb[n0 + r], bv = bib[n0 + r];
#pragma unroll
    for (int j = 0; j < 8; ++j) {
      float v1 = c1[j] + g;  v1 = v1 > 0.f ? v1 : 0.01f * v1;
      float v2 = c2[j] + bv; v2 = v2 > 0.f ? v2 : 0.01f * v2;
      outs[wave][j + 8 * h][n0 + r] = v1 + v2;
    }
  }
  __syncthreads();
  int t = threadIdx.x;
  if (t < 64) {
    int w2 = t >> 4, rr = t & 15;
    long grow = (long)blockIdx.x * 64 + t;
    if (grow < NTOT) {
      float ss = 0.f;
      for (int k2 = 0; k2 < 64; ++k2) { float v = outs[w2][rr][k2]; ss += v * v; }
      float inv = 1.f / fmaxf(sqrtf(ss), 1e-12f);
      float* ep = E    + (size_t)grow * 64;
      float* sp = Esum + (size_t)grow * 64;
      for (int k2 = 0; k2 < 64; ++k2) {
        float v = outs[w2][rr][k2] * inv;
        ep[k2] = v; sp[k2] += v;
      }
    }
  }
}

// ---------------- gather batch embeddings ----------------
__global__ void k_gather(const float* __restrict__ Esum, const int* __restrict__ uid,
                         const int* __restrict__ iid, float* ue, float* ie) {
  int b = blockIdx.x; int t = threadIdx.x;       // block = 64
  ue[(size_t)b * 64 + t] = Esum[(size_t)uid[b] * 64 + t];
  ie[(size_t)b * 64 + t] = Esum[((size_t)USERS + iid[b]) * 64 + t];
}

// ---------------- generic GEMM + bias + activation (WMMA f32 16x16x4) ----------------
// ACT: 0=none, 1=tanh. grid.x = M/128 (block 256 = 8 waves), grid.y = N/16.
// The 16-column weight slice for this block is staged transposed in LDS (one ds_load_b64/frag).
template<int K, int ACT>
__global__ void __launch_bounds__(256) k_gemm(
    const float* __restrict__ A, int lda,
    const float* __restrict__ Wm, int ldw,
    const float* __restrict__ bias,
    float* __restrict__ C, int ldc) {
  __shared__ float wT[16][K + 4];   // (K+4) % 64 is 4 for K in {64,256}, 36 for K=96 -> conflict-free
  int tid = threadIdx.x;
  int n0 = (int)blockIdx.y * 16;
  for (int i = tid; i < 16 * K; i += 256) {
    int k = i >> 4, n = i & 15;
    wT[n][k] = Wm[(size_t)k * ldw + n0 + n];
  }
  __syncthreads();

  int lane = tid & 31, wave = tid >> 5;
  int row0 = ((int)blockIdx.x * 8 + wave) * 16;
  int r = lane & 15, h = lane >> 4, kk = h * 2;
  const float* arow = A + (size_t)(row0 + r) * lda + kk;
  const float* wrow = &wT[r][kk];
  v8f c = {};
#pragma unroll
  for (int k0 = 0; k0 < K; k0 += 4) {
    v2f a = ld2(arow + k0);
    v2f b = ld2(wrow + k0);
    c = wmma4(a, b, c);
  }
  float bv = bias[n0 + r];
#pragma unroll
  for (int j = 0; j < 8; ++j) {
    float v = c[j] + bv;
    if (ACT == 1) v = tanhf(v);
    C[(size_t)(row0 + j + 8 * h) * ldc + (n0 + r)] = v;
  }
}

// ---------------- tiny self-attention over L=3 levels (one wave per batch item) ----------------
__global__ void k_attn(const float* __restrict__ recs,   // [3][B][32]
                       const float* __restrict__ qw, const float* __restrict__ qb,
                       const float* __restrict__ kw, const float* __restrict__ kb,
                       const float* __restrict__ vw, const float* __restrict__ vb,
                       float* att) {                     // [B][3][32]
  __shared__ float xs[8][3][33];
  int lane = threadIdx.x & 31, wave = threadIdx.x >> 5;
  int b = (int)blockIdx.x * 8 + wave;
  for (int s = 0; s < 3; ++s)
    xs[wave][s][lane] = recs[(size_t)s * BB * LEAF + (size_t)b * LEAF + lane];
  __syncthreads();
  float q[3], kv[3], vv[3];
  for (int s = 0; s < 3; ++s) { q[s] = qb[lane]; kv[s] = kb[lane]; vv[s] = vb[lane]; }
  for (int k2 = 0; k2 < 32; ++k2) {
    float wq = qw[k2 * 32 + lane], wk = kw[k2 * 32 + lane], wv = vw[k2 * 32 + lane];
#pragma unroll
    for (int s = 0; s < 3; ++s) {
      float xv = xs[wave][s][k2];
      q[s] += xv * wq; kv[s] += xv * wk; vv[s] += xv * wv;
    }
  }
  const float scale = 0.17677669529663687f;  // 1/sqrt(32)
  float aw[3][3];
  for (int s = 0; s < 3; ++s)
    for (int t = 0; t < 3; ++t)
      aw[s][t] = wave_sum32(q[s] * kv[t]) * scale;
  for (int s = 0; s < 3; ++s) {
    float m = fmaxf(aw[s][0], fmaxf(aw[s][1], aw[s][2]));
    float e0 = __expf(aw[s][0] - m), e1 = __expf(aw[s][1] - m), e2 = __expf(aw[s][2] - m);
    float inv = 1.f / (e0 + e1 + e2);
    aw[s][0] = e0 * inv; aw[s][1] = e1 * inv; aw[s][2] = e2 * inv;
  }
  for (int s = 0; s < 3; ++s) {
    float o = aw[s][0] * vv[0] + aw[s][1] * vv[1] + aw[s][2] * vv[2];
    att[(size_t)b * 96 + s * 32 + lane] = o;
  }
}

// ---------------- diag = (sigmoid(u_att)-sigmoid(i_att)) * skills ----------------
__global__ void k_diag(const float* __restrict__ uatt, const float* __restrict__ iatt,
                       const float* __restrict__ skills, float* diag) {
  long i = (long)blockIdx.x * blockDim.x + threadIdx.x;   // grid exact = B*96
  float a = 1.f / (1.f + __expf(-uatt[i]));
  float c = 1.f / (1.f + __expf(-iatt[i]));
  diag[i] = (a - c) * skills[i];
}

// ---------------- final 64->2 projection + softmax ----------------
__global__ void k_head(const float* __restrict__ h, const float* __restrict__ prw,
                       const float* __restrict__ prb, float* out) {
  int b = (int)blockIdx.x * blockDim.x + threadIdx.x;     // grid exact = B
  const float* hp = h + (size_t)b * 64;
  float l0 = prb[0], l1 = prb[1];
  for (int d = 0; d < 64; ++d) { float v = hp[d]; l0 += v * prw[d * 2]; l1 += v * prw[d * 2 + 1]; }
  float m = fmaxf(l0, l1);
  float e0 = __expf(l0 - m), e1 = __expf(l1 - m);
  float inv = 1.f / (e0 + e1);
  out[(size_t)b * 2] = e0 * inv; out[(size_t)b * 2 + 1] = e1 * inv;
}

// ---------------- contrastive: normalize + perturbed copies + pos ----------------
__global__ void k_cl_prep(const float* __restrict__ uatt, const float* __restrict__ iatt,
                          float* n1, float* n2, float* pos) {
  int lane = threadIdx.x & 31;
  int g = (int)blockIdx.x * 8 + (threadIdx.x >> 5);       // 0..6*B-1
  int sideI = g / (NLVL * BB);
  int rem = g - sideI * (NLVL * BB);
  int l = rem / BB;
  int b = rem - l * BB;
  const float* att = sideI ? iatt : uatt;
  float rv = att[(size_t)b * 96 + l * 32 + lane];
  float s1 = wave_sum32(rv * rv);
  float n1v = rv / fmaxf(sqrtf(s1), 1e-12f);
  float u = hash01((unsigned)(g * 32 + lane) * 2654435761u + 0x9E3779B9u);
  float sn = wave_sum32(u * u);
  float nl = u / fmaxf(sqrtf(sn), 1e-12f);
  float sg = (rv > 0.f) ? 1.f : (rv < 0.f ? -1.f : 0.f);
  float rn = rv + sg * nl * EPSV;
  float s2 = wave_sum32(rn * rn);
  float n2v = rn / fmaxf(sqrtf(s2), 1e-12f);
  n1[(size_t)g * 32 + lane] = n1v;
  n2[(size_t)g * 32 + lane] = n2v;
  float d = wave_sum32(n1v * n2v);
  if (lane == 0) pos[g] = __expf(d * (1.f / TAUV));
}

// ---------------- ttl[b] = sum_j exp(n1[b].n2[j]/tau) via WMMA A@B^T tiles ----------------
// block=128 (4 waves); task = group(6) x rowtile(256); one wave = 16 rows over all 4096 cols.
__global__ void __launch_bounds__(128) k_cl_ttl(
    const float* __restrict__ n1, const float* __restrict__ n2, float* ttl) {
  int lane = threadIdx.x & 31;
  int task = (int)blockIdx.x * 4 + (threadIdx.x >> 5);    // 0..1535
  int g = task >> 8;
  int row0 = (task & 255) * 16;
  int r = lane & 15, kk = (lane >> 4) * 2;
  const float* arow = n1 + (size_t)g * BB * LEAF + (size_t)(row0 + r) * LEAF + kk;
  const float* bbase = n2 + (size_t)g * BB * LEAF + kk;
  float acc[8];
#pragma unroll
  for (int j = 0; j < 8; ++j) acc[j] = 0.f;
  const float invT = 1.f / TAUV;
  for (int n0 = 0; n0 < BB; n0 += 16) {
    const float* brow = bbase + (size_t)(n0 + r) * LEAF;  // B = n2^T: frags load like A-frags
    v8f c = {};
#pragma unroll
    for (int k0 = 0; k0 < LEAF; k0 += 4) {
      v2f a = ld2(arow + k0);
      v2f b = ld2(brow + k0);
      c = wmma4(a, b, c);
    }
#pragma unroll
    for (int j = 0; j < 8; ++j) {
      float e = __expf(c[j] * invT);
      e += __shfl_xor(e, 1); e += __shfl_xor(e, 2);
      e += __shfl_xor(e, 4); e += __shfl_xor(e, 8);   // row-sum within 16-lane half
      acc[j] += e;
    }
  }
  if (r == 0) {
    int h = lane >> 4;
#pragma unroll
    for (int j = 0; j < 8; ++j)
      ttl[(size_t)g * BB + row0 + j + 8 * h] = acc[j];
  }
}

// ---------------- deterministic final reduction: cl = mean(-log(pos/ttl)) summed over sides ----------------
__global__ void k_cl_reduce(const float* __restrict__ pos, const float* __restrict__ ttl,
                            float* outcl) {
  __shared__ float red[256];
  float s = 0.f;
  for (int i = threadIdx.x; i < 6 * BB; i += 256)
    s += __logf(ttl[i]) - __logf(pos[i]);
  red[threadIdx.x] = s;
  __syncthreads();
  for (int st = 128; st > 0; st >>= 1) {
    if ((int)threadIdx.x < st) red[threadIdx.x] += red[threadIdx.x + st];
    __syncthreads();
  }
  if (threadIdx.x == 0) *outcl = red[0] * (1.f / (NLVL * (float)BB));
}

extern "C" void kernel_launch(void* const* d_in, const int* in_sizes, int n_in,
                              void* d_out, int out_size, void* d_ws, size_t ws_size,
                              hipStream_t stream) {
  const float* ego   = (const float*)d_in[0];
  const float* gc_w  = (const float*)d_in[1];
  const float* gc_b  = (const float*)d_in[2];
  const float* bi_w  = (const float*)d_in[3];
  const float* bi_b  = (const float*)d_in[4];
  const float* uh_w  = (const float*)d_in[5];
  const float* uh_b  = (const float*)d_in[6];
  const float* ih_w  = (const float*)d_in[7];
  const float* ih_b  = (const float*)d_in[8];
  const float* ud_w1 = (const float*)d_in[9];
  const float* ud_b1 = (const float*)d_in[10];
  const float* ud_w2 = (const float*)d_in[11];
  const float* ud_b2 = (const float*)d_in[12];
  const float* id_w1 = (const float*)d_in[13];
  const float* id_b1 = (const float*)d_in[14];
  const float* id_w2 = (const float*)d_in[15];
  const float* id_b2 = (const float*)d_in[16];
  const float* q_w   = (const float*)d_in[17];
  const float* q_b   = (const float*)d_in[18];
  const float* k_w   = (const float*)d_in[19];
  const float* k_b   = (const float*)d_in[20];
  const float* v_w   = (const float*)d_in[21];
  const float* v_b   = (const float*)d_in[22];
  const float* dg_w1 = (const float*)d_in[23];
  const float* dg_b1 = (const float*)d_in[24];
  const float* dg_w2 = (const float*)d_in[25];
  const float* dg_b2 = (const float*)d_in[26];
  const float* pr_w  = (const float*)d_in[27];
  const float* pr_b  = (const float*)d_in[28];
  const float* adj_vals = (const float*)d_in[29];
  const float* skills   = (const float*)d_in[30];
  const int* user_id  = (const int*)d_in[31];
  const int* item_id  = (const int*)d_in[32];
  const int* adj_rows = (const int*)d_in[33];
  const int* adj_cols = (const int*)d_in[34];
  float* out = (float*)d_out;

  float* W = (float*)d_ws;
  size_t o = 0;
  float* E     = W + o; o += (size_t)NPAD * DD;
  float* Esum  = W + o; o += (size_t)NPAD * DD;
  float* side  = W + o; o += (size_t)NPAD * DD;
  float* ue    = W + o; o += (size_t)BB * DD;
  float* ie    = W + o; o += (size_t)BB * DD;
  float* ureps = W + o; o += (size_t)NLVL * BB * DD;
  float* ireps = W + o; o += (size_t)NLVL * BB * DD;
  float* uhid  = W + o; o += (size_t)NLVL * BB * HIDN;
  float* ihid  = W + o; o += (size_t)NLVL * BB * HIDN;
  float* urecs = W + o; o += (size_t)NLVL * BB * LEAF;
  float* irecs = W + o; o += (size_t)NLVL * BB * LEAF;
  float* uatt  = W + o; o += (size_t)BB * NLVL * LEAF;
  float* iatt  = W + o; o += (size_t)BB * NLVL * LEAF;
  float* diag  = W + o; o += (size_t)BB * NLVL * LEAF;
  float* t1    = W + o; o += (size_t)BB * DD;
  float* h2    = W + o; o += (size_t)BB * DD;
  float* n1b   = W + o; o += (size_t)6 * BB * LEAF;
  float* n2b   = W + o; o += (size_t)6 * BB * LEAF;
  float* posb  = W + o; o += (size_t)6 * BB;
  float* ttlb  = W + o; o += (size_t)6 * BB;

  const long NPE = (long)NPAD * DD;            // 9,601,024 = 256 * 37504
  const int gInit = (int)(NPE / 256);

  // ---- NGCF propagation ----
  k_init_E<<<gInit, 256, 0, stream>>>(ego, E, Esum);
  for (int l = 0; l < 2; ++l) {
    k_zero<<<gInit, 256, 0, stream>>>(side, NPE);
    k_spmm<<<NNZE / 8, 256, 0, stream>>>(adj_vals, adj_rows, adj_cols, E, side);
    k_ngcf_dense<<<NPAD / 64, 128, 0, stream>>>(side, E, Esum,
        gc_w + (size_t)l * DD * DD, gc_b + (size_t)l * DD,
        bi_w + (size_t)l * DD * DD, bi_b + (size_t)l * DD);
  }
  k_gather<<<BB, 64, 0, stream>>>(Esum, user_id, item_id, ue, ie);

  // ---- per-level hierarchy projections + disentangled encoders ----
  for (int l = 0; l < NLVL; ++l) {
    // users
    k_gemm<64, 1><<<dim3(BB / 128, DD / 16), 256, 0, stream>>>(ue, DD,
        uh_w + (size_t)l * DD * DD, DD, uh_b + (size_t)l * DD,
        ureps + (size_t)l * BB * DD, DD);
    k_gemm<64, 1><<<dim3(BB / 128, HIDN / 16), 256, 0, stream>>>(ureps + (size_t)l * BB * DD, DD,
        ud_w1 + (size_t)l * DD * HIDN, HIDN, ud_b1 + (size_t)l * HIDN,
        uhid + (size_t)l * BB * HIDN, HIDN);
    k_gemm<256, 0><<<dim3(BB / 128, LEAF / 16), 256, 0, stream>>>(uhid + (size_t)l * BB * HIDN, HIDN,
        ud_w2 + (size_t)l * HIDN * LEAF, LEAF, ud_b2 + (size_t)l * LEAF,
        urecs + (size_t)l * BB * LEAF, LEAF);
    // items
    k_gemm<64, 1><<<dim3(BB / 128, DD / 16), 256, 0, stream>>>(ie, DD,
        ih_w + (size_t)l * DD * DD, DD, ih_b + (size_t)l * DD,
        ireps + (size_t)l * BB * DD, DD);
    k_gemm<64, 1><<<dim3(BB / 128, HIDN / 16), 256, 0, stream>>>(ireps + (size_t)l * BB * DD, DD,
        id_w1 + (size_t)l * DD * HIDN, HIDN, id_b1 + (size_t)l * HIDN,
        ihid + (size_t)l * BB * HIDN, HIDN);
    k_gemm<256, 0><<<dim3(BB / 128, LEAF / 16), 256, 0, stream>>>(ihid + (size_t)l * BB * HIDN, HIDN,
        id_w2 + (size_t)l * HIDN * LEAF, LEAF, id_b2 + (size_t)l * LEAF,
        irecs + (size_t)l * BB * LEAF, LEAF);
  }

  // ---- attention across levels ----
  k_attn<<<BB / 8, 256, 0, stream>>>(urecs, q_w, q_b, k_w, k_b, v_w, v_b, uatt);
  k_attn<<<BB / 8, 256, 0, stream>>>(irecs, q_w, q_b, k_w, k_b, v_w, v_b, iatt);

  // ---- diagnosis head ----
  k_diag<<<(BB * 96) / 256, 256, 0, stream>>>(uatt, iatt, skills, diag);
  k_gemm<96, 1><<<dim3(BB / 128, DD / 16), 256, 0, stream>>>(diag, NLVL * LEAF,
      dg_w1, DD, dg_b1, t1, DD);
  k_gemm<64, 0><<<dim3(BB / 128, DD / 16), 256, 0, stream>>>(t1, DD,
      dg_w2, DD, dg_b2, h2, DD);
  k_head<<<BB / 256, 256, 0, stream>>>(h2, pr_w, pr_b, out);

  // ---- contrastive loss (scalar output at out[B*2]) ----
  k_cl_prep<<<(6 * BB) / 8, 256, 0, stream>>>(uatt, iatt, n1b, n2b, posb);
  k_cl_ttl<<<(6 * (BB / 16)) / 4, 128, 0, stream>>>(n1b, n2b, ttlb);
  k_cl_reduce<<<1, 256, 0, stream>>>(posb, ttlb, out + (size_t)BB * 2);
}